// LenardJones_56075093017197
// MI455X (gfx1250) — compile-verified
//
#include <hip/hip_runtime.h>

// Lennard-Jones all-pairs forces on gfx1250 using V_WMMA_F32_16X16X4_F32.
//
// dp_i = sum_j s_ij (q_i - q_j) = q_i * (sum_j s_ij) - sum_j s_ij q_j
//   s_ij = -dV/dr / r = 96 * r^-2 * sr6 * (2*sr6 - 1)   (EPS=4, SIGMA=1)
//
// The WMMA computes, per 16x16 tile, D[m][n] = dot(q_i, q_j) - 0.5*|q_j|^2
// by packing the spare K=3 slot: A[m][3] = 1, B[3][n] = -0.5*|q_j|^2.
// Then d2 = |q_i|^2 - 2*D with a loop-invariant |q_i|^2 term.
// Accumulators (sum s, sum s*q_j) stay in the C/D lane layout -> no cross-lane
// work in the hot loop; one 16-lane xor-shuffle reduction at the end.
//
// The diagonal tile (jt == i0 - jc, scalar) is peeled out of the tile loop:
// loops A/B run a branch-free unmasked body; one masked call handles it.
// r^-2 uses v_rcp_f32 + 1 Newton step instead of the ~15-op IEEE division.

typedef float v2f __attribute__((ext_vector_type(2)));
typedef float v8f __attribute__((ext_vector_type(8)));

#define N_PART          4096
#define ROWS_PER_WAVE   16
#define WAVES_PER_BLOCK 8
#define ROWS_PER_BLOCK  (ROWS_PER_WAVE * WAVES_PER_BLOCK)  // 128
#define JTILE           16
#define JCHUNK          1024   // LDS stage: 1024 * float4 = 16 KB

template<bool MASKED>
__device__ __forceinline__ void lj_tile(const float4* __restrict__ sj, int jt,
                                        int col, int h, v2f a,
                                        const float (&nive)[8],
                                        float (&accS)[8], float (&accX)[8],
                                        float (&accY)[8], float (&accZ)[8])
{
    const float4 qj = sj[jt + col];
    // B-matrix (4x16 f32): lanes 0-15 -> (K0,K1)=(x,y);
    //                      lanes 16-31 -> (K2,K3)=(z, -0.5*|q_j|^2)
    v2f b;
    b.x = h ? qj.z : qj.x;
    b.y = h ? (-0.5f * qj.w) : qj.y;
    v8f c = {};
    // D[m][n] = dot(q_i, q_j) - 0.5*|q_j|^2
    v8f d = __builtin_amdgcn_wmma_f32_16x16x4_f32(
        false, a, false, b, (short)0, c, false, false);
#pragma unroll
    for (int v = 0; v < 8; ++v) {
        float d2 = fmaxf(nive[v] - 2.0f * d[v], 1e-12f);
        float inv = __builtin_amdgcn_rcpf(d2);
        inv = inv * (2.0f - d2 * inv);                 // Newton step
        const float sr6 = inv * inv * inv;
        float s = 96.0f * inv * sr6 * (2.0f * sr6 - 1.0f);
        if (MASKED) s = ((v + 8 * h) == col) ? 0.0f : s;  // zero diagonal
        accS[v] += s;
        accX[v] += s * qj.x;
        accY[v] += s * qj.y;
        accZ[v] += s * qj.z;
    }
}

__global__ __launch_bounds__(256)
void lj_force_wmma_kernel(const float* __restrict__ q,
                          float* __restrict__ dp_out,  // per-jslice slab base
                          int jlen)
{
    __shared__ float4 sj[JCHUNK];  // x, y, z, |q|^2 for staged j-particles

    const int tid  = threadIdx.x;
    const int lane = tid & 31;
    const int wave = tid >> 5;
    const int col  = lane & 15;       // N index within 16x16 tile
    const int h    = lane >> 4;       // 0: lanes 0-15, 1: lanes 16-31
    const int i0   = (blockIdx.x * WAVES_PER_BLOCK + wave) * ROWS_PER_WAVE;
    const int jstart = blockIdx.y * jlen;

    // Scalar (SGPR) copy of the wave-uniform row base for uniform control flow.
    const int i0_s = __builtin_amdgcn_readfirstlane(i0);

    // Each lane owns row (i0 + col)'s coordinates.
    const int irow_own = i0 + col;
    const float qx = q[irow_own * 3 + 0];
    const float qy = q[irow_own * 3 + 1];
    const float qz = q[irow_own * 3 + 2];
    const float ni_own = qx * qx + qy * qy + qz * qz;

    // A-matrix (16x4 f32): lanes 0-15 -> (K0,K1)=(x,y); lanes 16-31 -> (K2,K3)=(z,1)
    v2f a;
    a.x = h ? qz : qx;
    a.y = h ? 1.0f : qy;   // K3 = 1 so B's K3 row adds -0.5*|q_j|^2 to D

    // |q_i|^2 for the 8 rows this lane's D elements cover (M = v + 8*h).
    float nive[8];
#pragma unroll
    for (int v = 0; v < 8; ++v) nive[v] = __shfl(ni_own, v + 8 * h, 32);

    float accS[8], accX[8], accY[8], accZ[8];
#pragma unroll
    for (int v = 0; v < 8; ++v) { accS[v] = 0.f; accX[v] = 0.f; accY[v] = 0.f; accZ[v] = 0.f; }

    for (int jc = jstart; jc < jstart + jlen; jc += JCHUNK) {
        __syncthreads();
        // Stage JCHUNK j-particles (coords + norm) into LDS; uniform trip count.
        for (int t = tid; t < JCHUNK; t += 256) {
            const int j = jc + t;
            const float x = q[j * 3 + 0];
            const float y = q[j * 3 + 1];
            const float z = q[j * 3 + 2];
            sj[t] = make_float4(x, y, z, x * x + y * y + z * z);
        }
        __syncthreads();

        // Diagonal tile offset within this chunk (scalar; 16-aligned).
        const int jd = i0_s - jc;
        const bool hasDiag = (jd >= 0) && (jd < JCHUNK);
        const int boundA = hasDiag ? jd : JCHUNK;

        for (int jt = 0; jt < boundA; jt += JTILE)
            lj_tile<false>(sj, jt, col, h, a, nive, accS, accX, accY, accZ);

        if (hasDiag) {
            lj_tile<true>(sj, jd, col, h, a, nive, accS, accX, accY, accZ);
            for (int jt = jd + JTILE; jt < JCHUNK; jt += JTILE)
                lj_tile<false>(sj, jt, col, h, a, nive, accS, accX, accY, accZ);
        }
    }

    // Reduce each row's partial sums across its 16 column lanes.
#pragma unroll
    for (int v = 0; v < 8; ++v) {
#pragma unroll
        for (int m = 1; m < 16; m <<= 1) {
            accS[v] += __shfl_xor(accS[v], m, 32);
            accX[v] += __shfl_xor(accX[v], m, 32);
            accY[v] += __shfl_xor(accY[v], m, 32);
            accZ[v] += __shfl_xor(accZ[v], m, 32);
        }
    }

    // Lane 0 writes rows i0..i0+7, lane 16 writes rows i0+8..i0+15.
    if (col == 0) {
        float* outp = dp_out + (size_t)blockIdx.y * (N_PART * 3);
#pragma unroll
        for (int v = 0; v < 8; ++v) {
            const int irow = i0 + v + 8 * h;
            const float qix = q[irow * 3 + 0];
            const float qiy = q[irow * 3 + 1];
            const float qiz = q[irow * 3 + 2];
            outp[irow * 3 + 0] = qix * accS[v] - accX[v];
            outp[irow * 3 + 1] = qiy * accS[v] - accY[v];
            outp[irow * 3 + 2] = qiz * accS[v] - accZ[v];
        }
    }
}

__global__ void lj_reduce_kernel(const float* __restrict__ partials,
                                 float* __restrict__ dp, int jsplit)
{
    const int e = blockIdx.x * blockDim.x + threadIdx.x;
    if (e < N_PART * 3) {
        float s = 0.0f;
        for (int k = 0; k < jsplit; ++k) s += partials[(size_t)k * (N_PART * 3) + e];
        dp[e] = s;
    }
}

__global__ void lj_dq_kernel(const float* __restrict__ p,
                             const float* __restrict__ mass,
                             float* __restrict__ dq)
{
    const int e = blockIdx.x * blockDim.x + threadIdx.x;
    if (e < N_PART * 3) dq[e] = p[e] / mass[e / 3];
}

extern "C" void kernel_launch(void* const* d_in, const int* in_sizes, int n_in,
                              void* d_out, int out_size, void* d_ws, size_t ws_size,
                              hipStream_t stream) {
    const float* q    = (const float*)d_in[0];  // [4096,3]
    const float* p    = (const float*)d_in[1];  // [4096,3]
    const float* mass = (const float*)d_in[2];  // [4096,1]
    // d_in[3] = t (unused by the reference math)

    float* dq = (float*)d_out;                  // first 12288 floats
    float* dp = (float*)d_out + N_PART * 3;     // next 12288 floats

    // Split the j-range for occupancy (up to 2048 waves); deterministic
    // workspace partials + reduction (no float atomics). Fall back to
    // fewer slices if the workspace is too small.
    const size_t slab = (size_t)N_PART * 3 * sizeof(float);
    int jsplit = 1;
    if (ws_size >= 8 * slab)      jsplit = 8;
    else if (ws_size >= 4 * slab) jsplit = 4;
    const int jlen = N_PART / jsplit;

    float* target = (jsplit == 1) ? dp : (float*)d_ws;
    dim3 grid(N_PART / ROWS_PER_BLOCK, jsplit);
    lj_force_wmma_kernel<<<grid, 256, 0, stream>>>(q, target, jlen);

    if (jsplit > 1) {
        lj_reduce_kernel<<<(N_PART * 3 + 255) / 256, 256, 0, stream>>>(
            (const float*)d_ws, dp, jsplit);
    }
    lj_dq_kernel<<<(N_PART * 3 + 255) / 256, 256, 0, stream>>>(p, mass, dq);
}